// ContConv1d_29583734735380
// MI455X (gfx1250) — compile-verified
//
#include <hip/hip_runtime.h>

// Problem constants (from reference): BS=2, L=2048, CIN=32, COUT=32, HID=128, K=16
#define BS    2
#define LSEQ  2048
#define CIN_  32
#define COUT_ 32
#define HID_  128
#define KOFF  16
#define NBI   (BS * LSEQ)          // 4096 (b,i) groups
#define W3N   (CIN_ * COUT_)       // 1024

typedef float v2f __attribute__((ext_vector_type(2)));
typedef float v8f __attribute__((ext_vector_type(8)));

static __device__ __forceinline__ v8f wmma_f32_16x16x4(v2f a, v2f b, v8f c) {
  // 8-arg form: (neg_a, A, neg_b, B, c_mod, C, reuse_a, reuse_b)
  return __builtin_amdgcn_wmma_f32_16x16x4_f32(false, a, false, b, (short)0, c,
                                               false, false);
}

// ---------------------------------------------------------------------------
// Kernel 1: per (b,i) group (16 banded offsets k=1..16):
//   h1[k,h] = relu(dt_k * W1[h] + b1[h])          (layer 1, scalar input)
//   h2[k,:] = relu(h1[k,:] @ W2 + b2)             (WMMA GEMM, 16x128 @ 128x128)
//   Hsum[g,:] = sum_{k valid} h2[k,:]             (masked row reduction)
// One wave per group; 4 waves / block; 1024 blocks.
// ---------------------------------------------------------------------------
__global__ __launch_bounds__(128) void cc1d_h2sum_kernel(
    const float* __restrict__ times, const float* __restrict__ W1,
    const float* __restrict__ b1, const float* __restrict__ W2,
    const float* __restrict__ b2, float* __restrict__ Hsum) {
  __shared__ float sW1[HID_], sB1[HID_], sB2[HID_];
  const int tid = threadIdx.x;
  if (tid < HID_) {
    sW1[tid] = W1[tid];
    sB1[tid] = b1[tid];
    sB2[tid] = b2[tid];
  }
  __syncthreads();

  const int wave = tid >> 5;
  const int lane = tid & 31;
  const int g = blockIdx.x * 4 + wave;        // group index in [0, 4096)
  const int b = g >> 11;                       // / LSEQ
  const int i = g & (LSEQ - 1);

  // Each lane owns row m = lane&15 of the 16-row tile (k-offset = m+1).
  const int m = lane & 15;
  const int j = i - 1 - m;
  const float ti = times[b * LSEQ + i];
  float dtv = 0.0f;
  if (j >= 0) dtv = fmaxf(ti - times[b * LSEQ + j], 0.0f);

  const int khalf = (lane < 16) ? 0 : 2;       // K-half per ISA A/B layout

  const v8f vzero = {0.f, 0.f, 0.f, 0.f, 0.f, 0.f, 0.f, 0.f};
  v8f acc[8];
#pragma unroll
  for (int nt = 0; nt < 8; ++nt) acc[nt] = vzero;

  for (int kk = 0; kk < 32; ++kk) {            // K = 128 in steps of 4
    const int h0 = 4 * kk + khalf;
    v2f a;
    a.x = fmaxf(dtv * sW1[h0]     + sB1[h0],     0.0f);   // h1 on the fly
    a.y = fmaxf(dtv * sW1[h0 + 1] + sB1[h0 + 1], 0.0f);
#pragma unroll
    for (int nt = 0; nt < 8; ++nt) {           // 8 N-tiles of 16 -> 128 cols
      const int col = nt * 16 + m;
      v2f bb;
      bb.x = W2[h0 * HID_ + col];
      bb.y = W2[(h0 + 1) * HID_ + col];
      acc[nt] = wmma_f32_16x16x4(a, bb, acc[nt]);
    }
  }

  // Epilogue: +b2, relu, mask invalid rows (k-offset > i), reduce over rows.
  const int rbase = (lane < 16) ? 0 : 8;       // D-layout M base per lane half
#pragma unroll
  for (int nt = 0; nt < 8; ++nt) {
    const int col = nt * 16 + m;
    const float bias = sB2[col];
    float partial = 0.0f;
#pragma unroll
    for (int mm = 0; mm < 8; ++mm) {
      const int row = rbase + mm;              // k index 0..15 (offset row+1)
      const float h2 = fmaxf(acc[nt][mm] + bias, 0.0f);
      partial += (i >= row + 1) ? h2 : 0.0f;
    }
    partial += __shfl_xor(partial, 16, 32);    // combine the two lane halves
    if (lane < 16) Hsum[g * HID_ + col] = partial;
  }
}

// ---------------------------------------------------------------------------
// Kernel 2: per tile of 16 (b,i) rows:
//   KV = Hsum_tile(16x128) @ W3(128x1024)       (WMMA GEMM)
//   out[r,o] = sum_c f[r,c]*KV[r,c*32+o] + min(i_r,16)*sum_c f[r,c]*b3[c*32+o]
// 8 waves / block, each wave covers 8 of the 64 N-tiles; LDS atomics fuse the
// 32x32 feature contraction without materializing KV in global memory.
// ---------------------------------------------------------------------------
__global__ __launch_bounds__(256) void cc1d_out_kernel(
    const float* __restrict__ Hsum, const float* __restrict__ features,
    const float* __restrict__ W3, const float* __restrict__ b3,
    float* __restrict__ out) {
  __shared__ float sHs[16 * HID_];    // 8 KB
  __shared__ float sF[16 * CIN_];     // 2 KB
  __shared__ float sOut[16 * COUT_];  // 2 KB
  const int tid = threadIdx.x;
  const int g0 = blockIdx.x * 16;

  for (int t = tid; t < 16 * HID_; t += 256) sHs[t] = Hsum[g0 * HID_ + t];
  for (int t = tid; t < 16 * CIN_; t += 256) sF[t] = features[g0 * CIN_ + t];
  for (int t = tid; t < 16 * COUT_; t += 256) sOut[t] = 0.0f;
  __syncthreads();

  const int wave = tid >> 5;
  const int lane = tid & 31;
  const int m = lane & 15;
  const int khalf = (lane < 16) ? 0 : 2;

  const v8f vzero = {0.f, 0.f, 0.f, 0.f, 0.f, 0.f, 0.f, 0.f};
  v8f acc[8];
#pragma unroll
  for (int t = 0; t < 8; ++t) acc[t] = vzero;

  for (int kk = 0; kk < 32; ++kk) {
    const int h0 = 4 * kk + khalf;
    v2f a;
    a.x = sHs[m * HID_ + h0];
    a.y = sHs[m * HID_ + h0 + 1];
#pragma unroll
    for (int t = 0; t < 8; ++t) {
      const int nt = wave * 8 + t;             // N-tile 0..63
      const int col = nt * 16 + m;
      v2f bb;
      bb.x = W3[h0 * W3N + col];
      bb.y = W3[(h0 + 1) * W3N + col];
      acc[t] = wmma_f32_16x16x4(a, bb, acc[t]);
    }
  }

  // Scatter KV * f into the 16x32 output tile. col = c*32 + o.
  const int rbase = (lane < 16) ? 0 : 8;
#pragma unroll
  for (int t = 0; t < 8; ++t) {
    const int nt = wave * 8 + t;
    const int c = nt >> 1;
    const int o = ((nt & 1) << 4) | m;
#pragma unroll
    for (int mm = 0; mm < 8; ++mm) {
      const int row = rbase + mm;
      atomicAdd(&sOut[row * COUT_ + o], sF[row * CIN_ + c] * acc[t][mm]);
    }
  }
  __syncthreads();

  // Add nvalid * (f @ b3-matrix) and write out.
  for (int idx = tid; idx < 16 * COUT_; idx += 256) {
    const int r = idx >> 5;
    const int o = idx & 31;
    const int g = g0 + r;
    const int i = g & (LSEQ - 1);
    const float nval = (float)((i < KOFF) ? i : KOFF);
    float s = 0.0f;
#pragma unroll
    for (int c = 0; c < CIN_; ++c) s += sF[r * CIN_ + c] * b3[c * COUT_ + o];
    out[g * COUT_ + o] = sOut[idx] + nval * s;
  }
}

extern "C" void kernel_launch(void* const* d_in, const int* in_sizes, int n_in,
                              void* d_out, int out_size, void* d_ws,
                              size_t ws_size, hipStream_t stream) {
  const float* times    = (const float*)d_in[0];
  const float* features = (const float*)d_in[1];
  const float* W1       = (const float*)d_in[2];
  const float* b1       = (const float*)d_in[3];
  const float* W2       = (const float*)d_in[4];
  const float* b2       = (const float*)d_in[5];
  const float* W3       = (const float*)d_in[6];
  const float* b3       = (const float*)d_in[7];
  float* out  = (float*)d_out;
  float* Hsum = (float*)d_ws;  // 4096 * 128 floats = 2 MB scratch

  cc1d_h2sum_kernel<<<dim3(NBI / 4), dim3(128), 0, stream>>>(times, W1, b1, W2,
                                                             b2, Hsum);
  cc1d_out_kernel<<<dim3(NBI / 16), dim3(256), 0, stream>>>(Hsum, features, W3,
                                                            b3, out);
}